// TensorProduct_26663156973854
// MI455X (gfx1250) — compile-verified
//
#include <hip/hip_runtime.h>
#include <math.h>

typedef __attribute__((ext_vector_type(2))) float v2f;
typedef __attribute__((ext_vector_type(8))) float v8f;

#define KDIM 84      // K padded 81 -> 84 (21 WMMA steps of 4)
#define NPAD 64      // N padded 51 -> 64 (4 tiles of 16)
#define NCOL 51
#define WAVES 4
#define ROWS_PER_BLOCK 64   // 4 waves * 16 rows

// ---------------- instruction / column tables ----------------
__constant__ int c_l1[15] = {0,0,0,1,1,1,1,1,1,2,2,2,2,2,2};
__constant__ int c_l2[15] = {0,1,2,0,1,1,1,2,2,0,1,1,2,2,2};
__constant__ int c_l3[15] = {0,1,2,1,0,1,2,1,2,2,1,2,0,1,2};
__constant__ int c_colbase[15] = {0,1,4,9,12,13,16,21,24,29,34,37,42,43,46};

__constant__ int c_obase[NCOL] = {
  0, 32,33,34, 128,129,130,131,132, 288,289,290, 384, 416,417,418,
  512,513,514,515,516, 672,673,674, 768,769,770,771,772,
  928,929,930,931,932, 1088,1089,1090, 1184,1185,1186,1187,1188,
  1344, 1376,1377,1378, 1472,1473,1474,1475,1476 };
__constant__ int c_osz[NCOL] = {
  1, 3,3,3, 5,5,5,5,5, 3,3,3, 1, 3,3,3, 5,5,5,5,5, 3,3,3, 5,5,5,5,5,
  5,5,5,5,5, 3,3,3, 5,5,5,5,5, 1, 3,3,3, 5,5,5,5,5 };
__constant__ int c_wslot[NCOL] = {
  0, 1,1,1, 2,2,2,2,2, 3,3,3, 4, 5,5,5, 6,6,6,6,6, 7,7,7, 8,8,8,8,8,
  9,9,9,9,9, 10,10,10, 11,11,11,11,11, 12, 13,13,13, 14,14,14,14,14 };

// ---------------- init kernel: build constant matrix on device ----------------
struct cplx { double re, im; };
__device__ inline cplx cmul(cplx a, cplx b){
  cplx r; r.re = a.re*b.re - a.im*b.im; r.im = a.re*b.im + a.im*b.re; return r;
}
__device__ double dfact(int n){ double r=1.0; for(int i=2;i<=n;++i) r*=(double)i; return r; }

__device__ double su2cg(int j1,int j2,int j3,int m1,int m2,int m3){
  if(m3 != m1+m2) return 0.0;
  double pref = sqrt((2.0*j3+1.0)*dfact(j3+j1-j2)*dfact(j3-j1+j2)*dfact(j1+j2-j3)/dfact(j1+j2+j3+1));
  pref *= sqrt(dfact(j3+m3)*dfact(j3-m3)*dfact(j1-m1)*dfact(j1+m1)*dfact(j2-m2)*dfact(j2+m2));
  int kmin = 0;
  if(j2-j3-m1 > kmin) kmin = j2-j3-m1;
  if(j1-j3+m2 > kmin) kmin = j1-j3+m2;
  int kmax = j1+j2-j3;
  if(j1-m1 < kmax) kmax = j1-m1;
  if(j2+m2 < kmax) kmax = j2+m2;
  double s = 0.0;
  for(int k=kmin;k<=kmax;++k){
    double d = dfact(k)*dfact(j1+j2-j3-k)*dfact(j1-m1-k)*dfact(j2+m2-k)
             * dfact(j3-j2+m1+k)*dfact(j3-j1-m2+k);
    s += ((k&1)? -1.0 : 1.0)/d;
  }
  return pref*s;
}

__device__ void build_q(int l, cplx q[5][5]){
  for(int i=0;i<5;++i) for(int j=0;j<5;++j){ q[i][j].re=0.0; q[i][j].im=0.0; }
  const double is2 = 1.0/sqrt(2.0);
  for(int m=-l;m<0;++m){
    q[l+m][l-m].re = is2;          // q[l+m, l+|m|] = 1/sqrt(2)
    q[l+m][l+m].im = -is2;         // q[l+m, l-|m|] = -i/sqrt(2)
  }
  q[l][l].re = 1.0;
  for(int m=1;m<=l;++m){
    double s = (m&1)? -1.0 : 1.0;
    q[l+m][l+m].re = s*is2;        // q[l+m, l+m] = (-1)^m/sqrt(2)
    q[l+m][l-m].im = s*is2;        // q[l+m, l-m] = i*(-1)^m/sqrt(2)
  }
  cplx f; // (-i)^l
  if(l==0){ f.re=1.0; f.im=0.0; }
  else if(l==1){ f.re=0.0; f.im=-1.0; }
  else { f.re=-1.0; f.im=0.0; }
  for(int i=0;i<5;++i) for(int j=0;j<5;++j) q[i][j] = cmul(f, q[i][j]);
}

__global__ void init_C(float* __restrict__ Ct){
  int tid = threadIdx.x;
  for(int e=tid; e<NPAD*KDIM; e+=blockDim.x) Ct[e] = 0.f;
  __syncthreads();
  if(tid < 15){
    int l1 = c_l1[tid], l2 = c_l2[tid], l3 = c_l3[tid];
    cplx q1[5][5], q2[5][5], q3[5][5];
    build_q(l1,q1); build_q(l2,q2); build_q(l3,q3);
    double T[5][5][5];
    for(int i=0;i<5;++i)for(int j=0;j<5;++j)for(int k=0;k<5;++k) T[i][j][k]=0.0;
    for(int m1=-l1;m1<=l1;++m1)
      for(int m2=-l2;m2<=l2;++m2){
        int m3 = -(m1+m2);
        if(m3 >= -l3 && m3 <= l3){
          int p = l1 - l2 - m3;
          double sg = (p & 1) ? -1.0 : 1.0;
          T[l1+m1][l2+m2][l3+m3] = sg / sqrt(2.0*l3+1.0) * su2cg(l1,l2,l3,m1,m2,m1+m2);
        }
      }
    int n1=2*l1+1, n2=2*l2+1, n3=2*l3+1;
    double alpha = sqrt(2.0*l3+1.0);
    int ib = (l1==0)?0:((l1==1)?1:4);
    int jb = (l2==0)?0:((l2==1)?1:4);
    for(int a=0;a<n1;++a)
      for(int bb=0;bb<n2;++bb)
        for(int cc=0;cc<n3;++cc){
          double s = 0.0;
          for(int i=0;i<n1;++i)
            for(int j=0;j<n2;++j)
              for(int k=0;k<n3;++k){
                double t = T[i][j][k];
                if(t != 0.0){
                  cplx z = cmul(q1[i][a], q2[j][bb]);
                  z = cmul(z, q3[k][cc]);
                  s += t * z.re;   // T real -> only real part of q-product matters
                }
              }
          int ij  = (ib+a)*9 + (jb+bb);
          int col = c_colbase[tid] + cc;
          // stored n-major: Ct[col][k] so B-operand pairs are contiguous (ds_load_b64)
          Ct[col*KDIM + ij] = (float)(alpha * s);
        }
  }
}

// ---------------- main kernel: WMMA f32 GEMM + weighted scatter ----------------
__global__ __launch_bounds__(128) void tp_kernel(
    const float* __restrict__ x, const float* __restrict__ y,
    const float* __restrict__ W, float* __restrict__ out,
    const float* __restrict__ Ct, int R)
{
  __shared__ float Cs[NPAD*KDIM];            // 21504 B constant B-panel
  __shared__ float Ps[WAVES][16][KDIM];      // 21504 B per-wave A-panels
  __shared__ float Ws[WAVES][16][16];        //  4096 B per-row path weights

  const int tid  = threadIdx.x;
  for(int e=tid; e<NPAD*KDIM; e+=blockDim.x) Cs[e] = Ct[e];

  const int wave = tid >> 5;
  const int lane = tid & 31;
  const int m    = lane & 15;
  const int hi   = lane >> 4;
  const int rowbase = blockIdx.x * ROWS_PER_BLOCK + wave*16;

  // ---- build A-panel row m: P[e] = x[b,u,i]*y[b,j], e = i*9+j (pad 81..83 = 0)
  {
    int r  = rowbase + m;
    int rl = (r < R) ? r : (R-1);
    int b = rl >> 5, u = rl & 31;
    const float* xb = x + (long)b*288;
    const float* yb = y + (long)b*9;
    float xv[9], yv[9];
    xv[0] = xb[u];
#pragma unroll
    for(int t=0;t<3;++t) xv[1+t] = xb[32 + 3*u + t];
#pragma unroll
    for(int t=0;t<5;++t) xv[4+t] = xb[128 + 5*u + t];
#pragma unroll
    for(int j=0;j<9;++j) yv[j] = yb[j];

#pragma unroll
    for(int t=0;t<42;++t){               // two lanes split the 84 K-entries
      int e = hi*42 + t;
      float v = 0.f;
      if(e < 81){ int ig = e/9; int jg = e - 9*ig; v = xv[ig]*yv[jg]; }
      Ps[wave][m][e] = v;
    }
#pragma unroll
    for(int t=0;t<8;++t){                // stage the 15 path weights for row m
      int idx = hi*8 + t;
      if(idx < 15) Ws[wave][m][idx] = W[(long)b*480 + idx*32 + u];
    }
  }
  __syncthreads();

  // ---- GEMM: 21 K-steps x 4 N-tiles of V_WMMA_F32_16X16X4_F32
  v8f acc[4];
#pragma unroll
  for(int i=0;i<4;++i)
#pragma unroll
    for(int j=0;j<8;++j) acc[i][j] = 0.f;

  for(int kk=0; kk<21; ++kk){
    // A 16x4 f32: lane L -> a_v = A[L&15][2*(L>>4)+v]
    v2f a = *(const v2f*)&Ps[wave][m][4*kk + 2*hi];
#pragma unroll
    for(int nt=0; nt<4; ++nt){
      // B 4x16 f32 (assumed mirror of A): b_v = B[2*(L>>4)+v][L&15]
      v2f bm = *(const v2f*)&Cs[(nt*16 + m)*KDIM + 4*kk + 2*hi];
      acc[nt] = __builtin_amdgcn_wmma_f32_16x16x4_f32(
          false, a, false, bm, (short)0, acc[nt], false, false);
    }
  }

  // ---- epilogue: D vgpr v holds (M = 8*hi + v, N = m + 16*nt); scale by path weight
#pragma unroll
  for(int nt=0; nt<4; ++nt){
    int ncol = nt*16 + m;
    if(ncol < NCOL){
      int obase = c_obase[ncol];
      int osz   = c_osz[ncol];
      int wslot = c_wslot[ncol];
#pragma unroll
      for(int v=0; v<8; ++v){
        int r2 = rowbase + hi*8 + v;
        if(r2 < R){
          int b2 = r2 >> 5, u2 = r2 & 31;
          float wv = Ws[wave][hi*8 + v][wslot];
          out[(long)b2*1632 + obase + u2*osz] = acc[nt][v] * wv;
        }
      }
    }
  }
}

extern "C" void kernel_launch(void* const* d_in, const int* in_sizes, int n_in,
                              void* d_out, int out_size, void* d_ws, size_t ws_size,
                              hipStream_t stream) {
  const float* x = (const float*)d_in[0];
  const float* y = (const float*)d_in[1];
  const float* W = (const float*)d_in[2];
  float* out = (float*)d_out;
  float* Ct  = (float*)d_ws;     // 64*84 floats = 21504 B

  int B = in_sizes[0] / 288;
  int R = B * 32;
  init_C<<<1, 128, 0, stream>>>(Ct);
  int grid = (R + ROWS_PER_BLOCK - 1) / ROWS_PER_BLOCK;
  tp_kernel<<<grid, 128, 0, stream>>>(x, y, W, out, Ct, R);
}